// RobustConv_54898271978225
// MI455X (gfx1250) — compile-verified
//
#include <hip/hip_runtime.h>
#include <hip/hip_bf16.h>

// ---------------------------------------------------------------------------
// RobustConv for MI455X (gfx1250, wave32, WMMA)
//   mean = relu(feat @ W_mean); var = relu(feat @ W_var); att = exp(-var)
//   m_msg = mean*att*norm1[src-node];  v_msg = var*att*att*norm2[src-node]
//   mean_agg = segsum(m_msg[src] -> dst) * norm1;  var_agg likewise * norm2
// ---------------------------------------------------------------------------

#define N_NODES   100000
#define N_EDGES   1600000
#define IN_FEATS  256
#define OUT_FEATS 128

typedef __attribute__((ext_vector_type(2))) float v2f;
typedef __attribute__((ext_vector_type(8))) float v8f;

// ---- kernel 1: zero output accumulators + degree counters -----------------
__global__ void rc_init_kernel(float* __restrict__ out, float* __restrict__ cnt) {
    size_t i = (size_t)blockIdx.x * blockDim.x + threadIdx.x;
    const size_t nout = (size_t)2 * N_NODES * OUT_FEATS;
    if (i < nout) out[i] = 0.0f;
    if (i < N_NODES) cnt[i] = 0.0f;
}

// ---- kernel 2: in-degrees --------------------------------------------------
__global__ void rc_degree_kernel(const int* __restrict__ dst, float* __restrict__ cnt) {
    int e = blockIdx.x * blockDim.x + threadIdx.x;
    if (e < N_EDGES) atomicAdd(&cnt[dst[e]], 1.0f);
}

// ---- kernel 3: norms -------------------------------------------------------
__global__ void rc_norm_kernel(const float* __restrict__ cnt,
                               float* __restrict__ norm1, float* __restrict__ norm2) {
    int i = blockIdx.x * blockDim.x + threadIdx.x;
    if (i < N_NODES) {
        float d = fmaxf(cnt[i], 1.0f);
        norm1[i] = rsqrtf(d);
        norm2[i] = 1.0f / d;
    }
}

// ---- kernel 4: fused dual GEMM (WMMA f32 16x16x4) + message epilogue ------
// Block = 256 threads = 8 waves. Block b computes rows [16b, 16b+16).
// Wave w computes the 16x16 tile at columns [16w, 16w+16) for BOTH matrices.
__global__ void __launch_bounds__(256)
rc_gemm_msg_kernel(const float* __restrict__ feat,
                   const float* __restrict__ Wm,
                   const float* __restrict__ Wv,
                   const float* __restrict__ norm1,
                   const float* __restrict__ norm2,
                   float* __restrict__ m_msg,
                   float* __restrict__ v_msg) {
    const int lane = threadIdx.x & 31;
    const int wave = threadIdx.x >> 5;
    const int row0 = blockIdx.x * 16;        // 6250 blocks * 16 = 100000 exactly
    const int col0 = wave * 16;              // 8 waves * 16 = 128 exactly
    const int idx  = lane & 15;              // A row / B col inside the tile
    const int kh   = (lane >> 4) * 2;        // K-pair owned by this lane half

    const float* __restrict__ arow = feat + (size_t)(row0 + idx) * IN_FEATS;

    v8f accM = {};
    v8f accV = {};

    #pragma unroll 4
    for (int k = 0; k < IN_FEATS; k += 4) {
        const int kk = k + kh;
        // A fragment: 16x4 f32  (VGPR0: K=kh, VGPR1: K=kh+1 per ISA layout)
        v2f a;
        a.x = arow[kk];
        a.y = arow[kk + 1];
        // B fragments: 4x16 f32, lane = column, lane-half = K-half
        v2f bm, bv;
        bm.x = Wm[(size_t)kk * OUT_FEATS + col0 + idx];
        bm.y = Wm[(size_t)(kk + 1) * OUT_FEATS + col0 + idx];
        bv.x = Wv[(size_t)kk * OUT_FEATS + col0 + idx];
        bv.y = Wv[(size_t)(kk + 1) * OUT_FEATS + col0 + idx];
        // D = A*B + C   (8 args: neg_a, A, neg_b, B, c_mod, C, reuse_a, reuse_b)
        accM = __builtin_amdgcn_wmma_f32_16x16x4_f32(false, a, false, bm,
                                                     (short)0, accM, false, false);
        accV = __builtin_amdgcn_wmma_f32_16x16x4_f32(false, a, false, bv,
                                                     (short)0, accV, false, false);
    }

    // Epilogue: C/D layout — VGPR i: lanes 0-15 -> M=i, lanes 16-31 -> M=i+8
    const int rbase = row0 + (lane >> 4) * 8;
    const int gcol  = col0 + idx;
    #pragma unroll
    for (int i = 0; i < 8; ++i) {
        const int grow = rbase + i;
        float mean = accM[i];
        float var  = accV[i];
        mean = mean > 0.0f ? mean : 0.0f;     // relu
        var  = var  > 0.0f ? var  : 0.0f;     // relu
        const float att = __expf(-var);       // GAMMA = 1
        const float n1  = norm1[grow];
        const float n2  = norm2[grow];
        m_msg[(size_t)grow * OUT_FEATS + gcol] = mean * att * n1;
        v_msg[(size_t)grow * OUT_FEATS + gcol] = var * att * att * n2;
    }
}

// ---- kernel 5: edge scatter (1 wave / edge, 4 feats / lane, float4 gather) -
__global__ void __launch_bounds__(256)
rc_scatter_kernel(const int* __restrict__ src, const int* __restrict__ dst,
                  const float* __restrict__ m_msg, const float* __restrict__ v_msg,
                  float* __restrict__ mean_agg, float* __restrict__ var_agg) {
    const long long gid = (long long)blockIdx.x * blockDim.x + threadIdx.x;
    const long long e = gid >> 5;
    if (e >= N_EDGES) return;
    const int lane = (int)(gid & 31);
    const int s = src[e];
    const int d = dst[e];
    const float4 mv = ((const float4*)(m_msg + (size_t)s * OUT_FEATS))[lane];
    const float4 vv = ((const float4*)(v_msg + (size_t)s * OUT_FEATS))[lane];
    float* mo = mean_agg + (size_t)d * OUT_FEATS + lane * 4;
    float* vo = var_agg  + (size_t)d * OUT_FEATS + lane * 4;
    atomicAdd(mo + 0, mv.x);
    atomicAdd(mo + 1, mv.y);
    atomicAdd(mo + 2, mv.z);
    atomicAdd(mo + 3, mv.w);
    atomicAdd(vo + 0, vv.x);
    atomicAdd(vo + 1, vv.y);
    atomicAdd(vo + 2, vv.z);
    atomicAdd(vo + 3, vv.w);
}

// ---- kernel 6: scale aggregated sums by dst-node norms --------------------
__global__ void rc_finalize_kernel(float* __restrict__ mean_agg,
                                   float* __restrict__ var_agg,
                                   const float* __restrict__ norm1,
                                   const float* __restrict__ norm2) {
    size_t i = (size_t)blockIdx.x * blockDim.x + threadIdx.x;
    if (i < (size_t)N_NODES * OUT_FEATS) {
        const int node = (int)(i >> 7);      // / OUT_FEATS (=128)
        mean_agg[i] *= norm1[node];
        var_agg[i]  *= norm2[node];
    }
}

extern "C" void kernel_launch(void* const* d_in, const int* in_sizes, int n_in,
                              void* d_out, int out_size, void* d_ws, size_t ws_size,
                              hipStream_t stream) {
    const float* feat  = (const float*)d_in[0];   // [N, 256]
    const float* Wmean = (const float*)d_in[1];   // [256, 128]
    const float* Wvar  = (const float*)d_in[2];   // [256, 128]
    const int*   src   = (const int*)d_in[3];     // [E]
    const int*   dst   = (const int*)d_in[4];     // [E]

    float* mean_agg = (float*)d_out;                               // [N,128]
    float* var_agg  = mean_agg + (size_t)N_NODES * OUT_FEATS;      // [N,128]

    // workspace layout (floats): cnt[N] | norm1[N] | norm2[N] | m_msg[N*128] | v_msg[N*128]
    float* ws    = (float*)d_ws;
    float* cnt   = ws;
    float* norm1 = ws + N_NODES;
    float* norm2 = ws + 2 * (size_t)N_NODES;
    float* m_msg = ws + 3 * (size_t)N_NODES;                       // 16B-aligned offset
    float* v_msg = m_msg + (size_t)N_NODES * OUT_FEATS;

    const size_t nout = (size_t)2 * N_NODES * OUT_FEATS;
    rc_init_kernel<<<(int)((nout + 255) / 256), 256, 0, stream>>>(mean_agg, cnt);

    rc_degree_kernel<<<(N_EDGES + 255) / 256, 256, 0, stream>>>(dst, cnt);

    rc_norm_kernel<<<(N_NODES + 255) / 256, 256, 0, stream>>>(cnt, norm1, norm2);

    // 6250 row-blocks of 16 rows; 8 wave32s each covering all 128 columns
    rc_gemm_msg_kernel<<<N_NODES / 16, 256, 0, stream>>>(feat, Wmean, Wvar,
                                                         norm1, norm2, m_msg, v_msg);

    const long long scatter_threads = (long long)N_EDGES * 32;
    rc_scatter_kernel<<<(int)((scatter_threads + 255) / 256), 256, 0, stream>>>(
        src, dst, m_msg, v_msg, mean_agg, var_agg);

    rc_finalize_kernel<<<(int)(((size_t)N_NODES * OUT_FEATS + 255) / 256), 256, 0, stream>>>(
        mean_agg, var_agg, norm1, norm2);
}